// ChannelGroupAttention_8289286882016
// MI455X (gfx1250) — compile-verified
//
#include <hip/hip_runtime.h>

typedef __attribute__((ext_vector_type(2))) float v2f;
typedef __attribute__((ext_vector_type(8))) float v8f;

#define HW      3136      // 56*56
#define NB      32
#define NC      256
#define NG      8
#define GSZ     32        // channels per group
#define TPB     224       // 7 waves * 32, divides HW (3136 = 14*224)
#define NWAVES  7
#define PSTRIDE 33        // padded pixel stride in Y LDS tile (conflict-free)

__global__ __launch_bounds__(TPB)
void cga_group_wmma_kernel(const float* __restrict__ x,
                           const float* __restrict__ G,
                           float* __restrict__ out) {
    __shared__ float Ylds[NWAVES * NG * PSTRIDE];

    const int tid  = threadIdx.x;
    const int lane = tid & 31;
    const int wave = tid >> 5;
    const int b    = blockIdx.y;
    const int hw   = blockIdx.x * TPB + tid;          // exact cover, no bounds check

    const size_t xbase = (size_t)b * NC * HW + hw;

    // ---- B-matrix registers for V_WMMA_F32_16X16X4_F32 -------------------
    // B is 4x16 f32: VGPR0 lanes 0-15 = row k0   cols 0-15, lanes 16-31 = row k0+2
    //               VGPR1 lanes 0-15 = row k0+1,            lanes 16-31 = row k0+3
    // Columns 8..15 are zero padding (G is 8x8).
    const int col = lane & 15;
    const int khi = lane >> 4;           // selects rows (k0, k0+1) vs (k0+2, k0+3)
    const bool vcol = col < NG;
    const float b00 = vcol ? G[(0 + 2 * khi) * NG + col] : 0.0f; // k-half 0, rows 0/2
    const float b01 = vcol ? G[(1 + 2 * khi) * NG + col] : 0.0f; //           rows 1/3
    const float b10 = vcol ? G[(4 + 2 * khi) * NG + col] : 0.0f; // k-half 1, rows 4/6
    const float b11 = vcol ? G[(5 + 2 * khi) * NG + col] : 0.0f; //           rows 5/7

    // ---- Stage 1: per-pixel group sums (bandwidth-dominant part) ---------
    float acc[NG];
    #pragma unroll
    for (int g = 0; g < NG; ++g) acc[g] = 0.0f;

    #pragma unroll
    for (int ci = 0; ci < NC; ++ci)
        acc[ci >> 5] += x[xbase + (size_t)ci * HW];   // immediate offsets, coalesced

    // ---- Stage 2: Y[p][go] = sum_g S[p][g] * G[g][go] via 2x2 WMMAs ------
    // A is 16x4 f32: VGPR0 lanes 0-15 = (M=lane, K=k0), lanes 16-31 = (M=lane-16, K=k0+2)
    //               VGPR1:      K=k0+1                /        K=k0+3
    // Wave holds 32 pixels (pixel = lane): chunk0 = pixels 0..15, chunk1 = 16..31.
    const bool low = lane < 16;
    float s[NG];
    #pragma unroll
    for (int g = 0; g < NG; ++g)
        s[g] = __shfl(acc[g], lane ^ 16, 32);         // cross-half exchange

    v8f d0 = {};  // Y for pixels 0..15 of this wave
    v8f d1 = {};  // Y for pixels 16..31
    #pragma unroll
    for (int h = 0; h < 2; ++h) {
        const int k0 = 4 * h;
        v2f bm; bm.x = h ? b10 : b00; bm.y = h ? b11 : b01;

        v2f a0;   // chunk 0
        a0.x = low ? acc[k0]     : s[k0 + 2];
        a0.y = low ? acc[k0 + 1] : s[k0 + 3];
        d0 = __builtin_amdgcn_wmma_f32_16x16x4_f32(false, a0, false, bm,
                                                   (short)0, d0, false, false);
        v2f a1;   // chunk 1
        a1.x = low ? s[k0]       : acc[k0 + 2];
        a1.y = low ? s[k0 + 1]   : acc[k0 + 3];
        d1 = __builtin_amdgcn_wmma_f32_16x16x4_f32(false, a1, false, bm,
                                                   (short)0, d1, false, false);
    }

    // ---- Stage 3: redistribute Y so lane L owns Y[pixel=L][0..7] ---------
    // D layout: VGPR r -> lanes 0-15: (M=r, N=lane); lanes 16-31: (M=8+r, N=lane-16)
    float* yw = &Ylds[wave * NG * PSTRIDE];
    if (col < NG) {                                   // col == output group here
        const int prow = low ? 0 : 8;
        #pragma unroll
        for (int r = 0; r < 8; ++r) {
            yw[col * PSTRIDE + (prow + r)]      = d0[r];
            yw[col * PSTRIDE + (16 + prow + r)] = d1[r];
        }
    }
    __syncthreads();

    float y[NG];
    #pragma unroll
    for (int g = 0; g < NG; ++g)
        y[g] = yw[g * PSTRIDE + lane];                // conflict-free

    // ---- Stage 4: broadcast each group value to its 32 output channels ---
    const size_t obase = (size_t)b * NC * HW + hw;
    #pragma unroll
    for (int g = 0; g < NG; ++g) {
        const float v = y[g];
        #pragma unroll
        for (int j = 0; j < GSZ; ++j)
            out[obase + (size_t)(g * GSZ + j) * HW] = v;  // immediate offsets, coalesced
    }
}

extern "C" void kernel_launch(void* const* d_in, const int* in_sizes, int n_in,
                              void* d_out, int out_size, void* d_ws, size_t ws_size,
                              hipStream_t stream) {
    (void)in_sizes; (void)n_in; (void)out_size; (void)d_ws; (void)ws_size;
    const float* x = (const float*)d_in[0];   // [32, 256, 56, 56] fp32
    const float* G = (const float*)d_in[1];   // [8, 8] fp32
    float* out = (float*)d_out;               // [32, 256, 56, 56] fp32

    dim3 grid(HW / TPB, NB);                  // (14, 32)
    dim3 block(TPB);                          // 224 threads = 7 waves
    cga_group_wmma_kernel<<<grid, block, 0, stream>>>(x, G, out);
}